// _BidLSTM_29978871726695
// MI455X (gfx1250) — compile-verified
//
#include <hip/hip_runtime.h>
#include <math.h>

// Bidirectional LSTM for MI455X (gfx1250).
// Phase 1: xz = x @ W + b for both directions (parallel WMMA f32 GEMM -> d_ws).
//          Workspace layout: xz[dir][t][col(512)][b(32)] so that the 8 batch
//          rows of every 16x16 C-tile are contiguous -> b128 stores/loads.
// Phase 2: sequential scan, one workgroup per direction.
//   - R cached in LDS in pair-interleaved layout (b64 B-operand loads)
//   - h double-buffered in LDS -> single barrier per timestep
//   - c state in WMMA accumulator registers, gates fused in-register
//   - next step's xz slab prefetched into registers (hides HBM latency)
//
// Shapes: B=32, T=2048, C=128, U=128, 4U=512.
// d_ws assumed >= 2*T*B*512*4 = 256 MiB (xz for both directions, f32).

typedef __attribute__((ext_vector_type(2))) float v2f;
typedef __attribute__((ext_vector_type(8))) float v8f;

#define B_  32
#define T_  2048
#define C_  128
#define U_  128
#define G4_ 512   // 4*U

__device__ __forceinline__ v8f wmma_f32_16x16x4(v2f a, v2f b, v8f c) {
  // 8 args: (neg_a, A, neg_b, B, c_mod, C, reuse_a, reuse_b)
  return __builtin_amdgcn_wmma_f32_16x16x4_f32(
      false, a, false, b, (short)0, c, false, false);
}

// Fast sigmoid/tanh: v_exp_f32 + v_rcp_f32; saturate correctly at +/-inf.
__device__ __forceinline__ float fast_sigm(float x) {
  return __builtin_amdgcn_rcpf(1.0f + __expf(-x));
}
__device__ __forceinline__ float fast_tanh(float x) {
  return 1.0f - 2.0f * __builtin_amdgcn_rcpf(1.0f + __expf(2.0f * x));
}

// ---------------------------------------------------------------------------
// Kernel 1: xz[dir][t][col][b] = (x[b, (dir? T-1-t : t), :] @ W_dir + b_dir)[col]
// M dimension ordered as r' = t*32 + b: one 16-row tile = fixed t, 16 b-rows.
// grid = (4096 mtiles, 4, 2 dirs), block = 256 (8 waves, one 16x16 tile each)
// ---------------------------------------------------------------------------
__global__ void lstm_xz_gemm(const float* __restrict__ x,
                             const float* __restrict__ W_f,
                             const float* __restrict__ b_f,
                             const float* __restrict__ W_b,
                             const float* __restrict__ b_b,
                             float* __restrict__ xz) {
  const int lane  = threadIdx.x & 31;
  const int wave  = threadIdx.x >> 5;
  const int lrow  = lane & 15;
  const int khalf = lane >> 4;            // 0: K pair {k,k+1}; 1: {k+2,k+3}

  const int mtile = blockIdx.x;           // r' = t*32 + b ordering
  const int ntile = blockIdx.y * 8 + wave;// 0..31 over 512 cols
  const int dir   = blockIdx.z;

  const float* W    = dir ? W_b : W_f;
  const float* bias = dir ? b_b : b_f;

  const int col   = ntile * 16 + lrow;
  const int t     = mtile >> 1;           // fixed for the whole tile
  const int bhalf = mtile & 1;

  // A row for this lane: batch index varies across lanes, t fixed.
  const int b_lane = bhalf * 16 + lrow;
  const int src_t  = dir ? (T_ - 1 - t) : t;
  const float* xrow = x + ((size_t)b_lane * T_ + src_t) * C_;

  v8f acc;
  const float bv = bias[col];
#pragma unroll
  for (int j = 0; j < 8; ++j) acc[j] = bv;

#pragma unroll 8
  for (int k0 = 0; k0 < C_; k0 += 4) {
    const int kk = k0 + 2 * khalf;
    v2f a = *(const v2f*)&xrow[kk];                   // kk even -> 8B aligned
    v2f b;
    b.x = W[(size_t)kk * G4_ + col];
    b.y = W[(size_t)(kk + 1) * G4_ + col];
    acc = wmma_f32_16x16x4(a, b, acc);
  }

  // C-tile rows j are contiguous batch entries in [t][col][b] layout.
  const size_t base =
      ((size_t)(dir * T_ + t) * G4_ + col) * B_ + bhalf * 16 + 8 * khalf;
  float4 lo = make_float4(acc[0], acc[1], acc[2], acc[3]);
  float4 hi = make_float4(acc[4], acc[5], acc[6], acc[7]);
  *(float4*)&xz[base]     = lo;   // 32B-aligned
  *(float4*)&xz[base + 4] = hi;
}

// ---------------------------------------------------------------------------
// Kernel 2: the recurrence. grid = 2 (one block per direction), block = 512.
// Dynamic LDS: R pair-interleaved (256KB) + h double buffer (2x16KB) = 288KB.
// Wave (m, ucol) computes all four gate tiles for batch-rows [16m,16m+16) and
// u-columns [16*ucol, 16*ucol+16); K-loop outermost so one A load feeds four
// independent WMMA chains (ILP=4).
// ---------------------------------------------------------------------------
__global__ void lstm_scan(const float* __restrict__ xz,
                          const float* __restrict__ R_f,
                          const float* __restrict__ R_b,
                          float* __restrict__ out) {
  extern __shared__ float smem[];
  float* Rp = smem;                      // [U_/2][G4_][2] pair-interleaved
  float* hA = smem + U_ * G4_;           // [B_][U_]
  float* hB = hA + B_ * U_;              // [B_][U_]

  const int dir  = blockIdx.x;
  const int tid  = threadIdx.x;
  const float* R = dir ? R_b : R_f;

  // Stage R into LDS, interleaving adjacent K rows so B operands are b64:
  // Rp[((k>>1)*G4_ + col)*2 + (k&1)] = R[k*G4_ + col]
  for (int i = tid; i < U_ * G4_; i += blockDim.x) {
    const int k = i >> 9;          // G4_ = 512
    const int c = i & (G4_ - 1);
    Rp[(((k >> 1) * G4_ + c) << 1) + (k & 1)] = R[i];
  }
  for (int i = tid; i < B_ * U_; i += blockDim.x) hA[i] = 0.0f;
  __syncthreads();

  const int lane  = tid & 31;
  const int wave  = tid >> 5;       // 0..15
  const int lrow  = lane & 15;
  const int khalf = lane >> 4;
  const int m     = wave & 1;       // batch halftile
  const int ucol  = wave >> 1;      // 0..7: u-column block

  const int abase   = (m * 16 + lrow) * U_;  // A-tile row base in h LDS
  const int u       = ucol * 16 + lrow;
  const int rowbase = m * 16 + 8 * khalf;    // batch offset of C-layout rows

  v8f cst;                          // cell state, C-layout registers
#pragma unroll
  for (int j = 0; j < 8; ++j) cst[j] = 0.0f;

  const float* xz0 = xz + (size_t)dir * T_ * B_ * G4_;
  // Per-gate element offset within a [col][b] slab (32B-aligned).
  int goff[4];
#pragma unroll
  for (int g = 0; g < 4; ++g) goff[g] = (g * U_ + u) * B_ + rowbase;

  // Preload xz tiles for t = 0 (two b128 loads per gate).
  v8f xzr[4];
#pragma unroll
  for (int g = 0; g < 4; ++g) {
    float4 lo = *(const float4*)&xz0[goff[g]];
    float4 hi = *(const float4*)&xz0[goff[g] + 4];
    xzr[g][0] = lo.x; xzr[g][1] = lo.y; xzr[g][2] = lo.z; xzr[g][3] = lo.w;
    xzr[g][4] = hi.x; xzr[g][5] = hi.y; xzr[g][6] = hi.z; xzr[g][7] = hi.w;
  }

  float* hcur = hA;
  float* hnxt = hB;

  for (int t = 0; t < T_; ++t) {
    v8f zacc[4];
#pragma unroll
    for (int g = 0; g < 4; ++g) zacc[g] = xzr[g];

    // Prefetch next step's xz slab (overlaps with the WMMA work below).
    if (t + 1 < T_) {
      const float* p = xz0 + (size_t)(t + 1) * B_ * G4_;
#pragma unroll
      for (int g = 0; g < 4; ++g) {
        float4 lo = *(const float4*)&p[goff[g]];
        float4 hi = *(const float4*)&p[goff[g] + 4];
        xzr[g][0] = lo.x; xzr[g][1] = lo.y; xzr[g][2] = lo.z; xzr[g][3] = lo.w;
        xzr[g][4] = hi.x; xzr[g][5] = hi.y; xzr[g][6] = hi.z; xzr[g][7] = hi.w;
      }
    }

    // z += h @ R : one A load per k-step feeds 4 independent WMMA chains.
#pragma unroll 8
    for (int k0 = 0; k0 < U_; k0 += 4) {
      const int kk = k0 + 2 * khalf;           // even
      const v2f a = *(const v2f*)&hcur[abase + kk];
#pragma unroll
      for (int g = 0; g < 4; ++g) {
        const v2f b = *(const v2f*)&Rp[((kk >> 1) * G4_ + g * U_ + u) << 1];
        zacc[g] = wmma_f32_16x16x4(a, b, zacc[g]);
      }
    }

    // Gates fused in-register; h written to the *other* LDS buffer.
    const int tt = dir ? (T_ - 1 - t) : t;
#pragma unroll
    for (int j = 0; j < 8; ++j) {
      const int brow = rowbase + j;            // batch index (C-layout)
      const float iv = fast_sigm(zacc[0][j]);
      const float fv = fast_sigm(zacc[1][j]);
      const float gv = fast_tanh(zacc[2][j]);
      const float ov = fast_sigm(zacc[3][j]);
      const float cn = fv * cst[j] + iv * gv;
      cst[j] = cn;
      const float hn = ov * fast_tanh(cn);
      hnxt[brow * U_ + u] = hn;
      // sequence_output[b, tt, dir*128 + u]
      out[((size_t)brow * T_ + tt) * (2 * U_) + dir * U_ + u] = hn;
    }

    __syncthreads();   // single barrier: h double-buffered
    float* tmp = hcur; hcur = hnxt; hnxt = tmp;
  }

  // state_output = concat(h_f, c_f): forward direction only.
  if (dir == 0) {
    float* st = out + (size_t)B_ * T_ * (2 * U_);
#pragma unroll
    for (int j = 0; j < 8; ++j) {
      const int brow = rowbase + j;
      st[brow * (2 * U_) + u]      = hcur[brow * U_ + u];
      st[brow * (2 * U_) + U_ + u] = cst[j];
    }
  }
}

// ---------------------------------------------------------------------------
extern "C" void kernel_launch(void* const* d_in, const int* in_sizes, int n_in,
                              void* d_out, int out_size, void* d_ws, size_t ws_size,
                              hipStream_t stream) {
  const float* x   = (const float*)d_in[0];
  const float* W_f = (const float*)d_in[1];
  const float* R_f = (const float*)d_in[2];
  const float* b_f = (const float*)d_in[3];
  const float* W_b = (const float*)d_in[4];
  const float* R_b = (const float*)d_in[5];
  const float* b_b = (const float*)d_in[6];
  float* out = (float*)d_out;
  float* xz  = (float*)d_ws;   // 2 * T * B * 512 floats = 256 MiB

  // Phase 1: input projections for both directions.
  dim3 g1((B_ * T_) / 16, G4_ / (16 * 8), 2);
  lstm_xz_gemm<<<g1, 256, 0, stream>>>(x, W_f, b_f, W_b, b_b, xz);

  // Phase 2: the scan. 288KB dynamic LDS per block (CDNA5: 320KB/WGP).
  const int shmem = (U_ * G4_ + 2 * B_ * U_) * (int)sizeof(float);
  (void)hipFuncSetAttribute((const void*)lstm_scan,
                            hipFuncAttributeMaxDynamicSharedMemorySize, shmem);
  lstm_scan<<<dim3(2), 512, shmem, stream>>>(xz, R_f, R_b, out);
}